// PVRCNNPlusPlusVoxelization_730144440344
// MI455X (gfx1250) — compile-verified
//
#include <hip/hip_runtime.h>
#include <hip/hip_bf16.h>

// ---------------- problem constants (from reference) ----------------
#define NXG 1408
#define NYG 1600
#define NZG 40
#define NCELLS   (NXG * NYG * NZG)      // 90,112,000
#define NWORDS   (NCELLS / 32)          // 2,816,000
#define CHUNK_W  64                     // bitmap words per scan chunk
#define NCHUNKS  (NWORDS / CHUNK_W)     // 44,000
#define MAX_VOX  16000
#define MAX_PTS  32
#define NFEAT    64                     // 3 xyz + 61 feats
#define NF4      16                     // 64 floats = 16 float4

// ---------------- voxel key (replicates reference float math) ----------------
__device__ __forceinline__ int voxel_key(float x, float y, float z,
                                         int* cx, int* cy, int* cz) {
    const float x0 = 0.0f,   y0 = -40.0f, z0 = -3.0f;
    const float x1 = 70.4f,  y1 =  40.0f, z1 =  1.0f;
    const float vx = 0.05f,  vy = 0.05f,  vz = 0.1f;
    if (!(x >= x0 && x < x1 && y >= y0 && y < y1 && z >= z0 && z < z1))
        return -1;
    int ix = (int)floorf((x - x0) / vx);
    int iy = (int)floorf((y - y0) / vy);
    int iz = (int)floorf((z - z0) / vz);
    if (ix < 0 || ix >= NXG || iy < 0 || iy >= NYG || iz < 0 || iz >= NZG)
        return -1;
    *cx = ix; *cy = iy; *cz = iz;
    return iz * (NXG * NYG) + iy * NXG + ix;
}

// ---------------- K1: mark occupancy bitmap ----------------
__global__ void k_mark(const float* __restrict__ pf, unsigned* __restrict__ bitmap, int n) {
    int i = blockIdx.x * blockDim.x + threadIdx.x;
    if (i >= n) return;
    float x = pf[(size_t)i * NFEAT + 0];
    float y = pf[(size_t)i * NFEAT + 1];
    float z = pf[(size_t)i * NFEAT + 2];
    int cx, cy, cz;
    int key = voxel_key(x, y, z, &cx, &cy, &cz);
    if (key < 0) return;
    atomicOr(&bitmap[key >> 5], 1u << (key & 31));
}

// ---------------- K2: per-chunk popcount sums (b128 bitmap reads) ----------------
__global__ void k_chunksum(const uint4* __restrict__ bitmap4, unsigned* __restrict__ sums) {
    int c = blockIdx.x * blockDim.x + threadIdx.x;
    if (c >= NCHUNKS) return;
    const uint4* w = bitmap4 + (size_t)c * (CHUNK_W / 4);   // 16 x uint4 per chunk
    unsigned s = 0;
#pragma unroll 4
    for (int j = 0; j < CHUNK_W / 4; ++j) {
        uint4 v = w[j];
        s += __popc(v.x) + __popc(v.y) + __popc(v.z) + __popc(v.w);
    }
    sums[c] = s;
}

// ---------------- K3: single-block exclusive scan over chunk sums ----------------
__global__ void k_scan(const unsigned* __restrict__ sums, unsigned* __restrict__ prefix) {
    __shared__ unsigned tile[1024];
    __shared__ unsigned carry;
    if (threadIdx.x == 0) carry = 0;
    __syncthreads();
    for (int base = 0; base < NCHUNKS; base += 1024) {
        int i = base + (int)threadIdx.x;
        unsigned v = (i < NCHUNKS) ? sums[i] : 0u;
        tile[threadIdx.x] = v;
        __syncthreads();
        for (int off = 1; off < 1024; off <<= 1) {
            unsigned t = (threadIdx.x >= (unsigned)off) ? tile[threadIdx.x - off] : 0u;
            __syncthreads();
            tile[threadIdx.x] += t;
            __syncthreads();
        }
        if (i < NCHUNKS) prefix[i] = carry + tile[threadIdx.x] - v;  // exclusive
        __syncthreads();
        if (threadIdx.x == 0) carry += tile[1023];
        __syncthreads();
    }
}

// ---------------- K4: assign points to voxel slots ----------------
__global__ void k_assign(const float* __restrict__ pf,
                         const unsigned* __restrict__ bitmap,
                         const unsigned* __restrict__ prefix,
                         unsigned* __restrict__ counts,
                         int* __restrict__ dense,
                         int* __restrict__ coords_out,
                         int n) {
    int i = blockIdx.x * blockDim.x + threadIdx.x;
    if (i >= n) return;
    float x = pf[(size_t)i * NFEAT + 0];
    float y = pf[(size_t)i * NFEAT + 1];
    float z = pf[(size_t)i * NFEAT + 2];
    int cx, cy, cz;
    int key = voxel_key(x, y, z, &cx, &cy, &cz);
    if (key < 0) return;
    int word  = key >> 5;
    int bit   = key & 31;
    int chunk = word >> 6;                    // /CHUNK_W
    unsigned vid = prefix[chunk];
    for (int w = chunk << 6; w < word; ++w) vid += __popc(bitmap[w]);
    vid += __popc(bitmap[word] & ((1u << bit) - 1u));
    if (vid >= MAX_VOX) return;
    unsigned slot = atomicAdd(&counts[vid], 1u);
    if (slot < MAX_PTS) dense[vid * MAX_PTS + slot] = i;
    // coords identical for every point of the voxel -> idempotent plain stores
    coords_out[vid * 3 + 0] = cz;   // reversed (z,y,x)
    coords_out[vid * 3 + 1] = cy;
    coords_out[vid * 3 + 2] = cx;
}

// ---------------- K5: per-voxel ascending sort (recovers stable-sort rank) ----------------
__global__ void k_finalize(const unsigned* __restrict__ counts,
                           int* __restrict__ dense,
                           int* __restrict__ coords_out,
                           int* __restrict__ nump_out) {
    int v = blockIdx.x * blockDim.x + threadIdx.x;
    if (v >= MAX_VOX) return;
    unsigned c  = counts[v];
    int cc      = c < MAX_PTS ? (int)c : MAX_PTS;
    int* d = dense + v * MAX_PTS;
    for (int a = 1; a < cc; ++a) {          // counts are tiny (avg ~1)
        int val = d[a];
        int b = a - 1;
        while (b >= 0 && d[b] > val) { d[b + 1] = d[b]; --b; }
        d[b + 1] = val;
    }
    nump_out[v] = cc;
    if (c == 0) {
        coords_out[v * 3 + 0] = 0;
        coords_out[v * 3 + 1] = 0;
        coords_out[v * 3 + 2] = 0;
    }
}

// ---------------- K6: bandwidth gather via async global<->LDS DMA path ----------------
// Each thread moves TWO float4s: both async b128 loads issued before a single
// s_wait_asynccnt, doubling per-wave bytes in flight (1 KB/wave).
// 256 threads/block -> 512 float4 = 32 output rows per block.
// Grid is an exact multiple so EXEC is all-ones, no tail.
__global__ void k_gather(const float4* __restrict__ in4,
                         const int* __restrict__ dense,
                         float4* __restrict__ out4,
                         int n) {
    __shared__ __align__(16) float4 stage[512];
    const int tid = (int)threadIdx.x;
    const int e0  = blockIdx.x * 512 + tid;        // element index, pass 0
    const int e1  = e0 + 256;                      // element index, pass 1

    // pass 0 source
    int r0   = e0 >> 4;
    int q0   = e0 & 15;
    int idx0 = dense[r0] + 1;                      // reference: clip(dense+1, 0, n-1)
    idx0 = idx0 < 0 ? 0 : idx0;
    idx0 = idx0 > n - 1 ? n - 1 : idx0;
    const float4* g0 = in4 + (size_t)idx0 * NF4 + q0;

    // pass 1 source
    int r1   = e1 >> 4;
    int q1   = e1 & 15;
    int idx1 = dense[r1] + 1;
    idx1 = idx1 < 0 ? 0 : idx1;
    idx1 = idx1 > n - 1 ? n - 1 : idx1;
    const float4* g1 = in4 + (size_t)idx1 * NF4 + q1;

    float4* o0 = out4 + (size_t)e0;
    float4* o1 = out4 + (size_t)e1;

#if defined(__gfx1250__)
    // LDS byte address = low 32 bits of the generic shared-space pointer
    unsigned lds0 = (unsigned)(unsigned long long)(&stage[tid]);
    unsigned lds1 = (unsigned)(unsigned long long)(&stage[tid + 256]);
    // two async gathers in flight per wave before waiting
    asm volatile("global_load_async_to_lds_b128 %0, %1, off"
                 :: "v"(lds0), "v"(g0) : "memory");
    asm volatile("global_load_async_to_lds_b128 %0, %1, off"
                 :: "v"(lds1), "v"(g1) : "memory");
    asm volatile("s_wait_asynccnt 0" ::: "memory");
    asm volatile("global_store_async_from_lds_b128 %0, %1, off"
                 :: "v"(o0), "v"(lds0) : "memory");
    asm volatile("global_store_async_from_lds_b128 %0, %1, off"
                 :: "v"(o1), "v"(lds1) : "memory");
    // S_ENDPGM performs an implicit wait-idle; async stores drain before retire.
#else
    *o0 = *g0;
    *o1 = *g1;
#endif
}

// ---------------- host launch ----------------
extern "C" void kernel_launch(void* const* d_in, const int* in_sizes, int n_in,
                              void* d_out, int out_size, void* d_ws, size_t ws_size,
                              hipStream_t stream) {
    const float* pf = (const float*)d_in[0];
    const int n = in_sizes[0] / NFEAT;   // 400,000

    // workspace layout (bytes), total ~13.73 MB
    char* ws = (char*)d_ws;
    unsigned* bitmap  = (unsigned*)(ws);                                   // 11,264,000 B
    unsigned* sums    = (unsigned*)(ws + (size_t)NWORDS * 4);              //    176,000 B
    unsigned* prefix  = (unsigned*)(ws + (size_t)NWORDS * 4 + NCHUNKS * 4);//    176,000 B
    unsigned* counts  = (unsigned*)(ws + (size_t)NWORDS * 4 + 2ull * NCHUNKS * 4);               // 64,000 B
    int*      dense   = (int*)     (ws + (size_t)NWORDS * 4 + 2ull * NCHUNKS * 4 + MAX_VOX * 4); // 2,048,000 B

    // output layout (flat concat, element offsets)
    float* out_voxels = (float*)d_out;                                       // 16000*32*64
    int*   coords_out = (int*)((float*)d_out + (size_t)MAX_VOX * MAX_PTS * NFEAT);
    int*   nump_out   = (int*)((float*)d_out + (size_t)MAX_VOX * MAX_PTS * NFEAT + MAX_VOX * 3);

    // per-call init (graph-capture-safe async memsets)
    hipMemsetAsync(bitmap, 0x00, (size_t)NWORDS * 4, stream);
    hipMemsetAsync(counts, 0x00, (size_t)MAX_VOX * 4, stream);
    hipMemsetAsync(dense,  0xFF, (size_t)MAX_VOX * MAX_PTS * 4, stream);   // -1

    const int B = 256;
    k_mark    <<<(n + B - 1) / B, B, 0, stream>>>(pf, bitmap, n);
    k_chunksum<<<(NCHUNKS + B - 1) / B, B, 0, stream>>>((const uint4*)bitmap, sums);
    k_scan    <<<1, 1024, 0, stream>>>(sums, prefix);
    k_assign  <<<(n + B - 1) / B, B, 0, stream>>>(pf, bitmap, prefix, counts, dense, coords_out, n);
    k_finalize<<<(MAX_VOX + B - 1) / B, B, 0, stream>>>(counts, dense, coords_out, nump_out);

    // gather: 16000*32 rows * 16 float4 = 8,192,000 elements; 512/block -> 16,000 blocks, no tail
    const int total_f4 = MAX_VOX * MAX_PTS * NF4;
    k_gather<<<total_f4 / 512, 256, 0, stream>>>((const float4*)pf, dense, (float4*)out_voxels, n);
}